// EMN_58317065945275
// MI455X (gfx1250) — compile-verified
//
#include <hip/hip_runtime.h>
#include <math.h>

typedef __attribute__((ext_vector_type(2))) float v2f;
typedef __attribute__((ext_vector_type(8))) float v8f;

#define BB 64      // batch
#define SS 50      // sentences
#define LL 20      // words per sentence
#define QQ 20      // query words
#define DD 128     // embed dim
#define VV 32000   // vocab

// ---------------------------------------------------------------------------
// Stage 1: a_embed[b,s,:] = sum_l embedA[story[b,s,l]]   (grid=3200, block=128)
// ---------------------------------------------------------------------------
__global__ void k_a_embed(const int* __restrict__ story,
                          const float* __restrict__ embedA,
                          float* __restrict__ a_embed) {
  const int bs = blockIdx.x;       // 0..3199  (b*50+s)
  const int d  = threadIdx.x;      // 0..127
  const int* w = story + bs * LL;
  float acc = 0.f;
#pragma unroll
  for (int l = 0; l < LL; ++l)
    acc += embedA[(long)w[l] * DD + d];
  a_embed[bs * DD + d] = acc;
}

// Stage 1b: u[b,:] = sum_q embedB[query[b,q]]            (grid=64, block=128)
__global__ void k_u(const int* __restrict__ query,
                    const float* __restrict__ embedB,
                    float* __restrict__ u) {
  const int b = blockIdx.x;
  const int d = threadIdx.x;
  const int* w = query + b * QQ;
  float acc = 0.f;
#pragma unroll
  for (int q = 0; q < QQ; ++q)
    acc += embedB[(long)w[q] * DD + d];
  u[b * DD + d] = acc;
}

// ---------------------------------------------------------------------------
// Stage 2: ip[s] = sum_{b,k} a_embed[b,s,k]*u[b,k]       (grid=50, block=256)
// ---------------------------------------------------------------------------
__global__ void k_ip(const float* __restrict__ a_embed,
                     const float* __restrict__ u,
                     float* __restrict__ ip) {
  const int s = blockIdx.x;
  __shared__ float red[256];
  float acc = 0.f;
  for (int i = threadIdx.x; i < BB * DD; i += 256) {
    const int b = i >> 7;          // i / 128
    const int k = i & (DD - 1);
    acc += a_embed[(b * SS + s) * DD + k] * u[b * DD + k];
  }
  red[threadIdx.x] = acc;
  __syncthreads();
  for (int off = 128; off > 0; off >>= 1) {
    if (threadIdx.x < off) red[threadIdx.x] += red[threadIdx.x + off];
    __syncthreads();
  }
  if (threadIdx.x == 0) ip[s] = red[0];
}

// Stage 3: p = softmax(ip) over 50                       (grid=1, block=64)
__global__ void k_softmax_p(const float* __restrict__ ip, float* __restrict__ p) {
  __shared__ float red[64];
  const int t = threadIdx.x;
  const float v = (t < SS) ? ip[t] : -INFINITY;
  red[t] = v; __syncthreads();
  for (int off = 32; off > 0; off >>= 1) {
    if (t < off) red[t] = fmaxf(red[t], red[t + off]);
    __syncthreads();
  }
  const float mx = red[0];
  __syncthreads();
  const float e = (t < SS) ? expf(v - mx) : 0.f;
  red[t] = e; __syncthreads();
  for (int off = 32; off > 0; off >>= 1) {
    if (t < off) red[t] += red[t + off];
    __syncthreads();
  }
  if (t < SS) p[t] = e / red[0];
}

// Stage 4: ten3[b,:] = u[b,:] + sum_s a_embed[b,s,:]*p[s]  (grid=64, block=128)
__global__ void k_ten3(const float* __restrict__ a_embed,
                       const float* __restrict__ u,
                       const float* __restrict__ p,
                       float* __restrict__ ten3) {
  const int b = blockIdx.x;
  const int d = threadIdx.x;
  float acc = u[b * DD + d];
  for (int s = 0; s < SS; ++s)
    acc += a_embed[(b * SS + s) * DD + d] * p[s];
  ten3[b * DD + d] = acc;
}

// ---------------------------------------------------------------------------
// Stage 5: out = ten3 @ W^T + bias  via V_WMMA_F32_16X16X4_F32
// One wave per block; each wave owns a 16(M) x 64(N) strip = 4 C tiles
// sharing one A fragment per k-step. grid = (VV/64, BB/16), block = 32.
//
// A (16x4 f32): lanes 0-15 -> rows M=lane, K={k,k+1}; lanes 16-31 -> K={k+2,k+3}
// B (4x16 f32): lanes 0-15 -> cols N=lane, K={k,k+1}; lanes 16-31 -> K={k+2,k+3}
// C/D (16x16 f32): VGPR r -> row m0 + r + 8*(lane>=16), col n0 + (lane&15)
// ---------------------------------------------------------------------------
__global__ void __launch_bounds__(32)
k_gemm_wmma(const float* __restrict__ ten3,
            const float* __restrict__ Wm,
            const float* __restrict__ bias,
            float* __restrict__ out) {
  const int lane = threadIdx.x;            // 0..31, EXEC all ones
  const int m0   = blockIdx.y << 4;        // 0,16,32,48
  const int n0   = blockIdx.x << 6;        // multiple of 64
  const int half = lane >> 4;              // 0 or 1
  const int l15  = lane & 15;

  const float* __restrict__ arow = ten3 + (m0 + l15) * DD;
  const float* __restrict__ b0r  = Wm + (long)(n0 +  0 + l15) * DD;
  const float* __restrict__ b1r  = Wm + (long)(n0 + 16 + l15) * DD;
  const float* __restrict__ b2r  = Wm + (long)(n0 + 32 + l15) * DD;
  const float* __restrict__ b3r  = Wm + (long)(n0 + 48 + l15) * DD;

  v8f c0 = {}, c1 = {}, c2 = {}, c3 = {};

#pragma unroll 4
  for (int k = 0; k < DD; k += 4) {
    const int kk = k + half * 2;           // this half-wave's K pair
    const v2f a  = *(const v2f*)(arow + kk);
    const v2f b0 = *(const v2f*)(b0r  + kk);
    const v2f b1 = *(const v2f*)(b1r  + kk);
    const v2f b2 = *(const v2f*)(b2r  + kk);
    const v2f b3 = *(const v2f*)(b3r  + kk);
    c0 = __builtin_amdgcn_wmma_f32_16x16x4_f32(false, a, false, b0, (short)0, c0, false, false);
    c1 = __builtin_amdgcn_wmma_f32_16x16x4_f32(false, a, false, b1, (short)0, c1, false, false);
    c2 = __builtin_amdgcn_wmma_f32_16x16x4_f32(false, a, false, b2, (short)0, c2, false, false);
    c3 = __builtin_amdgcn_wmma_f32_16x16x4_f32(false, a, false, b3, (short)0, c3, false, false);
  }

  const int ncol = n0 + l15;
#pragma unroll
  for (int r = 0; r < 8; ++r) {
    const long m = m0 + r + 8 * half;
    out[m * VV + ncol +  0] = c0[r] + bias[ncol +  0];
    out[m * VV + ncol + 16] = c1[r] + bias[ncol + 16];
    out[m * VV + ncol + 32] = c2[r] + bias[ncol + 32];
    out[m * VV + ncol + 48] = c3[r] + bias[ncol + 48];
  }
}

// ---------------------------------------------------------------------------
// Stage 6: in-place row softmax over 32000               (grid=64, block=256)
// ---------------------------------------------------------------------------
__global__ void k_softmax_rows(float* __restrict__ out) {
  const int row = blockIdx.x;
  float* __restrict__ x = out + (long)row * VV;
  __shared__ float red[256];
  const int t = threadIdx.x;

  float m = -INFINITY;
  for (int i = t; i < VV; i += 256) m = fmaxf(m, x[i]);
  red[t] = m; __syncthreads();
  for (int off = 128; off > 0; off >>= 1) {
    if (t < off) red[t] = fmaxf(red[t], red[t + off]);
    __syncthreads();
  }
  const float mx = red[0];
  __syncthreads();

  float s = 0.f;
  for (int i = t; i < VV; i += 256) s += expf(x[i] - mx);
  red[t] = s; __syncthreads();
  for (int off = 128; off > 0; off >>= 1) {
    if (t < off) red[t] += red[t + off];
    __syncthreads();
  }
  const float inv = 1.0f / red[0];

  for (int i = t; i < VV; i += 256) x[i] = expf(x[i] - mx) * inv;
}

// ---------------------------------------------------------------------------
extern "C" void kernel_launch(void* const* d_in, const int* in_sizes, int n_in,
                              void* d_out, int out_size, void* d_ws, size_t ws_size,
                              hipStream_t stream) {
  const int*   story  = (const int*)d_in[0];     // [64,50,20]
  const int*   query  = (const int*)d_in[1];     // [64,20]
  const float* embedA = (const float*)d_in[2];   // [32000,128]
  const float* embedB = (const float*)d_in[3];   // [32000,128]
  const float* Wm     = (const float*)d_in[4];   // [32000,128]
  const float* bias   = (const float*)d_in[5];   // [32000]
  float* out = (float*)d_out;                    // [64,32000]
  float* ws  = (float*)d_ws;

  // workspace layout (floats), all offsets 64-float aligned
  float* a_embed = ws;                 // 3200*128 = 409600
  float* u       = ws + 409600;        // 64*128   = 8192
  float* ip      = ws + 417792;        // 64
  float* p       = ws + 417856;        // 64
  float* ten3    = ws + 417920;        // 64*128   = 8192

  k_a_embed  <<<BB * SS, DD, 0, stream>>>(story, embedA, a_embed);
  k_u        <<<BB,      DD, 0, stream>>>(query, embedB, u);
  k_ip       <<<SS,     256, 0, stream>>>(a_embed, u, ip);
  k_softmax_p<<<1,       64, 0, stream>>>(ip, p);
  k_ten3     <<<BB,      DD, 0, stream>>>(a_embed, u, p, ten3);

  dim3 gg(VV / 64, BB / 16);           // (500, 4) waves of 32
  k_gemm_wmma<<<gg, 32, 0, stream>>>(ten3, Wm, bias, out);

  k_softmax_rows<<<BB, 256, 0, stream>>>(out);
}